// GCN_56495999811736
// MI455X (gfx1250) — compile-verified
//
#include <hip/hip_runtime.h>

// ---------------------------------------------------------------------------
// GCN on MI455X (gfx1250). f32 end-to-end; dense 16x16 lin layer via
// V_WMMA_F32_16X16X4_F32 (exact); scatter via global_atomic_add_f32 which
// lands in the 192MB L2 (all node arrays ~70MB are L2-resident).
// ---------------------------------------------------------------------------

typedef __attribute__((ext_vector_type(2))) float v2f;
typedef __attribute__((ext_vector_type(8))) float v8f;

#define DFEAT 16

// ---- edge pass 1: weighted degree --------------------------------------
__global__ void k_deg(const int* __restrict__ ei, const int* __restrict__ ea,
                      const float* __restrict__ eemb, float* __restrict__ deg,
                      long E) {
  long stride = (long)gridDim.x * blockDim.x;
  for (long e = (long)blockIdx.x * blockDim.x + threadIdx.x; e < E; e += stride) {
    int c = ei[E + e];
    float w = eemb[ea[e]];
    atomicAdd(&deg[c], w);
  }
}

// ---- dinv = rsqrt(deg + 1)  (deg+1 >= 1 so always valid) ----------------
__global__ void k_dinv(float* __restrict__ deg, long n) {
  long i = (long)blockIdx.x * blockDim.x + threadIdx.x;
  if (i < n) deg[i] = rsqrtf(deg[i] + 1.0f);
}

// ---- embedding gather + xlin1 = h @ W1^T via WMMA f32 16x16x4 -----------
// one wave per 16-node tile; 8 waves (256 thr) per block -> 128 nodes/block
__global__ void k_embed_gemm(const int* __restrict__ xidx,
                             const float* __restrict__ embw,
                             const float* __restrict__ W1,
                             float* __restrict__ xlin1, long n) {
  __shared__ float sh[8][16][17];           // stride-17 pad vs bank conflicts
  const int wave = threadIdx.x >> 5;
  const int lane = threadIdx.x & 31;
  const long tileBase = ((long)blockIdx.x * 8 + wave) * 16;

  // stage 16 node embeddings (16x16 f32) into LDS: 8 floats per lane
  {
    const int row = lane & 15;
    const int colBase = (lane >> 4) * 8;
    const long node = tileBase + row;
    float4 v0 = {0.f, 0.f, 0.f, 0.f}, v1 = {0.f, 0.f, 0.f, 0.f};
    if (node < n) {
      const int idx = xidx[node];
      const float* src = embw + (size_t)idx * DFEAT + colBase;
      v0 = *(const float4*)(src);
      v1 = *(const float4*)(src + 4);
    }
    sh[wave][row][colBase + 0] = v0.x; sh[wave][row][colBase + 1] = v0.y;
    sh[wave][row][colBase + 2] = v0.z; sh[wave][row][colBase + 3] = v0.w;
    sh[wave][row][colBase + 4] = v1.x; sh[wave][row][colBase + 5] = v1.y;
    sh[wave][row][colBase + 6] = v1.z; sh[wave][row][colBase + 7] = v1.w;
  }
  __syncthreads();

  // A-fragment: lane L: M = L&15, Khalf = L>>4; VGPR j holds K = k0+2*Khalf+j
  // B = W1^T  (B[k][nn] = W1[nn*16+k]); same K striping, N = lane&15
  const int m     = lane & 15;
  const int khalf = lane >> 4;
  v8f acc = {};
#pragma unroll
  for (int ki = 0; ki < 4; ++ki) {
    const int k0 = ki * 4 + khalf * 2;
    v2f a, b;
    a.x = sh[wave][m][k0 + 0];
    a.y = sh[wave][m][k0 + 1];
    b.x = W1[m * DFEAT + k0 + 0];   // nn == m == lane&15
    b.y = W1[m * DFEAT + k0 + 1];
    acc = __builtin_amdgcn_wmma_f32_16x16x4_f32(
        /*neg_a=*/false, a, /*neg_b=*/false, b,
        /*c_mod=*/(short)0, acc, /*reuse_a=*/false, /*reuse_b=*/false);
  }

  // D layout: lane L: N = L&15; VGPR v holds M = v + 8*(L>>4)
  const int nn       = lane & 15;
  const int mhalf    = lane >> 4;
  const long nodeBase = tileBase + 8 * mhalf;
  if (tileBase + 16 <= n) {
    // fast path: full tile, one uniform branch, clause of 8 offset stores
    float* dst = xlin1 + (size_t)nodeBase * DFEAT + nn;
#pragma unroll
    for (int v = 0; v < 8; ++v) dst[v * DFEAT] = acc[v];
  } else {
    // tail tile only: per-row guard
#pragma unroll
    for (int v = 0; v < 8; ++v) {
      const long node = nodeBase + v;
      if (node < n) xlin1[(size_t)node * DFEAT + nn] = acc[v];
    }
  }
}

// ---- edge scatter, layer 1 (16 floats per edge) -------------------------
__global__ void k_scatter16(const int* __restrict__ ei, const int* __restrict__ ea,
                            const float* __restrict__ eemb,
                            const float* __restrict__ dinv,
                            const float* __restrict__ xlin1,
                            float* __restrict__ agg1, long E) {
  long stride = (long)gridDim.x * blockDim.x;
  for (long e = (long)blockIdx.x * blockDim.x + threadIdx.x; e < E; e += stride) {
    long en = e + stride;
    if (en < E) __builtin_prefetch(&xlin1[(size_t)ei[en] * DFEAT], 0, 1);
    const int r = ei[e];
    const int c = ei[E + e];
    const float w = eemb[ea[e]];
    const float norm = dinv[r] * w * dinv[c];
    const float4* xr = (const float4*)(xlin1 + (size_t)r * DFEAT);
    const float4 x0 = xr[0], x1 = xr[1], x2 = xr[2], x3 = xr[3];
    float* ac = agg1 + (size_t)c * DFEAT;
    atomicAdd(ac +  0, norm * x0.x); atomicAdd(ac +  1, norm * x0.y);
    atomicAdd(ac +  2, norm * x0.z); atomicAdd(ac +  3, norm * x0.w);
    atomicAdd(ac +  4, norm * x1.x); atomicAdd(ac +  5, norm * x1.y);
    atomicAdd(ac +  6, norm * x1.z); atomicAdd(ac +  7, norm * x1.w);
    atomicAdd(ac +  8, norm * x2.x); atomicAdd(ac +  9, norm * x2.y);
    atomicAdd(ac + 10, norm * x2.z); atomicAdd(ac + 11, norm * x2.w);
    atomicAdd(ac + 12, norm * x3.x); atomicAdd(ac + 13, norm * x3.y);
    atomicAdd(ac + 14, norm * x3.z); atomicAdd(ac + 15, norm * x3.w);
  }
}

// ---- h1 = relu(agg1 + dinv^2*xlin1 + b1);  xlin2 = h1 . W2 --------------
__global__ void k_layer2(const float* __restrict__ agg1,
                         const float* __restrict__ xlin1,
                         const float* __restrict__ dinv,
                         const float* __restrict__ b1,
                         const float* __restrict__ W2,
                         float* __restrict__ xlin2, long n) {
  long i = (long)blockIdx.x * blockDim.x + threadIdx.x;
  if (i >= n) return;
  const float di2 = dinv[i] * dinv[i];
  float acc = 0.f;
#pragma unroll
  for (int k = 0; k < DFEAT; ++k) {
    float h = agg1[i * DFEAT + k] + di2 * xlin1[i * DFEAT + k] + b1[k];
    h = fmaxf(h, 0.f);
    acc += h * W2[k];
  }
  xlin2[i] = acc;
}

// ---- edge scatter, layer 2 (scalar) -------------------------------------
__global__ void k_scatter1(const int* __restrict__ ei, const int* __restrict__ ea,
                           const float* __restrict__ eemb,
                           const float* __restrict__ dinv,
                           const float* __restrict__ xlin2,
                           float* __restrict__ agg2, long E) {
  long stride = (long)gridDim.x * blockDim.x;
  for (long e = (long)blockIdx.x * blockDim.x + threadIdx.x; e < E; e += stride) {
    long en = e + stride;
    if (en < E) __builtin_prefetch(&xlin2[(size_t)ei[en]], 0, 1);
    const int r = ei[e];
    const int c = ei[E + e];
    const float w = eemb[ea[e]];
    atomicAdd(&agg2[c], dinv[r] * w * dinv[c] * xlin2[r]);
  }
}

// ---- final: out = sum_i (agg2[i] + dinv^2*xlin2[i] + b2) -----------------
__global__ void k_final(const float* __restrict__ agg2,
                        const float* __restrict__ xlin2,
                        const float* __restrict__ dinv,
                        const float* __restrict__ b2,
                        float* __restrict__ out, long n) {
  __shared__ float red[256];
  long i = (long)blockIdx.x * blockDim.x + threadIdx.x;
  float val = 0.f;
  if (i < n) {
    const float di2 = dinv[i] * dinv[i];
    val = agg2[i] + di2 * xlin2[i] + b2[0];
  }
  red[threadIdx.x] = val;
  __syncthreads();
  for (int s = 128; s > 0; s >>= 1) {
    if ((int)threadIdx.x < s) red[threadIdx.x] += red[threadIdx.x + s];
    __syncthreads();
  }
  if (threadIdx.x == 0) atomicAdd(out, red[0]);
}

extern "C" void kernel_launch(void* const* d_in, const int* in_sizes, int n_in,
                              void* d_out, int out_size, void* d_ws, size_t ws_size,
                              hipStream_t stream) {
  const int*   xidx  = (const int*)  d_in[0];  // (N,1) int32
  const int*   eidx  = (const int*)  d_in[1];  // (2,E) int32
  const int*   eattr = (const int*)  d_in[2];  // (E,)  int32
  const float* embw  = (const float*)d_in[3];  // (128,16)
  const float* eembw = (const float*)d_in[4];  // (16,1)
  const float* W1    = (const float*)d_in[5];  // (16,16)
  const float* b1    = (const float*)d_in[6];  // (16,)
  const float* W2    = (const float*)d_in[7];  // (1,16)
  const float* b2    = (const float*)d_in[8];  // (1,)
  float* out = (float*)d_out;                  // (1,) f32

  const long n = in_sizes[0];
  const long E = in_sizes[2];
  const size_t nAl = (size_t)((n + 63) & ~63L);   // 256B-aligned regions

  float* ws    = (float*)d_ws;                 // 35*nAl floats (~70MB) used
  float* dinv  = ws;                           // [0, nAl)  deg -> dinv inplace
  float* xlin1 = ws + nAl;                     // 16*nAl
  float* agg1  = ws + 17 * nAl;                // 16*nAl
  float* xlin2 = ws + 33 * nAl;                // nAl
  float* agg2  = ws + 34 * nAl;                // nAl

  hipMemsetAsync(dinv, 0, (size_t)n * sizeof(float), stream);
  hipMemsetAsync(agg1, 0, (size_t)n * DFEAT * sizeof(float), stream);
  hipMemsetAsync(agg2, 0, (size_t)n * sizeof(float), stream);
  hipMemsetAsync(out,  0, sizeof(float), stream);

  const int  TB = 256;
  long eg = (E + TB - 1) / TB; if (eg > 8192) eg = 8192;   // grid-stride
  const long nb = (n + TB - 1) / TB;
  const long gb = (n + 127) / 128;                         // 128 nodes/block

  k_deg       <<<eg, TB, 0, stream>>>(eidx, eattr, eembw, dinv, E);
  k_dinv      <<<nb, TB, 0, stream>>>(dinv, n);
  k_embed_gemm<<<gb, TB, 0, stream>>>(xidx, embw, W1, xlin1, n);
  k_scatter16 <<<eg, TB, 0, stream>>>(eidx, eattr, eembw, dinv, xlin1, agg1, E);
  k_layer2    <<<nb, TB, 0, stream>>>(agg1, xlin1, dinv, b1, W2, xlin2, n);
  k_scatter1  <<<eg, TB, 0, stream>>>(eidx, eattr, eembw, dinv, xlin2, agg2, E);
  k_final     <<<nb, TB, 0, stream>>>(agg2, xlin2, dinv, b2, out, n);
}